// CausalityChainModel_54631984005498
// MI455X (gfx1250) — compile-verified
//
#include <hip/hip_runtime.h>
#include <math.h>
#include <stdint.h>

// ---- problem constants (match reference) ----
#define SIZE  64
#define NS    16384
#define LAT   128
#define NOISE 64
#define HID   256
#define BTR   2048
#define NIND  8192
#define BN_EPS 1e-5f
#define LRELU  0.01f

typedef __attribute__((ext_vector_type(2))) float v2f;
typedef __attribute__((ext_vector_type(8))) float v8f;
typedef unsigned int uint32;
typedef __attribute__((ext_vector_type(4))) uint32 v4u;
typedef __attribute__((ext_vector_type(8))) uint32 v8u;

// -----------------------------------------------------------------------------
// WMMA helpers: D = A(16xK, row-major) * B^T(NxK row-major -> B is KxN),
// using V_WMMA_F32_16X16X4_F32, wave32, one wave per 16x16 tile.
// A layout: lane L: M = L%16, K = 2*(L/16) + reg.  B: N = L%16, K = 2*(L/16)+reg.
// D layout: lane L: N = L%16, M = reg + 8*(L/16).
// -----------------------------------------------------------------------------
__device__ __forceinline__ void wmma_store_tile(float* Cp, int ldc, v8f d) {
  int lane = threadIdx.x & 31;
  int hlf = lane >> 4, lid = lane & 15;
#pragma unroll
  for (int r = 0; r < 8; ++r)
    Cp[(r + 8 * hlf) * ldc + lid] = d[r];
}

// ---------------------------------------------------------------------------
// 1) C = sigmoid(L - L^T) * (1-I);  loss_trans = sum((C@C) * C^T)
// ---------------------------------------------------------------------------
__global__ void k_conn(const float* __restrict__ L, float* __restrict__ C,
                       float* __restrict__ loss_trans) {
  __shared__ float sC[64 * 64];
  __shared__ float sh[256];
  for (int i = threadIdx.x; i < 4096; i += 256) {
    int r = i >> 6, c = i & 63;
    float v = 1.f / (1.f + expf(-(L[r * 64 + c] - L[c * 64 + r])));
    if (r == c) v = 0.f;
    sC[i] = v;
    C[i] = v;
  }
  __syncthreads();
  float acc = 0.f;
  for (int p = threadIdx.x; p < 4096; p += 256) {
    int i = p >> 6, j = p & 63;
    float d = 0.f;
    for (int k = 0; k < 64; ++k) d += sC[i * 64 + k] * sC[k * 64 + j];
    acc += d * sC[j * 64 + i];
  }
  sh[threadIdx.x] = acc;
  __syncthreads();
  for (int ofs = 128; ofs > 0; ofs >>= 1) {
    if (threadIdx.x < ofs) sh[threadIdx.x] += sh[threadIdx.x + ofs];
    __syncthreads();
  }
  if (threadIdx.x == 0) *loss_trans = sh[0];
}

// ---------------------------------------------------------------------------
// 2) H = X @ W^T   (M x K) * (N x K)^T, grid (M/16, N/16), 1 wave per block
// ---------------------------------------------------------------------------
__global__ void k_gemm_xwT(const float* __restrict__ X, const float* __restrict__ W,
                           float* __restrict__ H, int M, int N, int K) {
  int rt = blockIdx.x * 16, ct = blockIdx.y * 16;
  int lane = threadIdx.x & 31, hlf = lane >> 4, lid = lane & 15;
  const float* Arow = X + (rt + lid) * K;
  const float* Brow = W + (ct + lid) * K;
  v8f acc = {};
  for (int k = 0; k < K; k += 4) {
    int ka = k + 2 * hlf;
    v2f a, b;
    a.x = Arow[ka]; a.y = Arow[ka + 1];
    b.x = Brow[ka]; b.y = Brow[ka + 1];
    acc = __builtin_amdgcn_wmma_f32_16x16x4_f32(false, a, false, b, (short)0, acc,
                                                false, false);
  }
  wmma_store_tile(H + rt * N + ct, N, acc);
  (void)M;
}

// ---------------------------------------------------------------------------
// 3) per-column mean/biased-var over M rows (also used as column-mean helper)
// ---------------------------------------------------------------------------
__global__ void k_bn_stats(const float* __restrict__ H, float* __restrict__ mu,
                           float* __restrict__ var, int M, int N) {
  int n = blockIdx.x;
  float s = 0.f, s2 = 0.f;
  for (int r = threadIdx.x; r < M; r += blockDim.x) {
    float v = H[r * N + n];
    s += v;
    s2 += v * v;
  }
  __shared__ float sh[256], sh2[256];
  sh[threadIdx.x] = s; sh2[threadIdx.x] = s2;
  __syncthreads();
  for (int ofs = 128; ofs > 0; ofs >>= 1) {
    if (threadIdx.x < ofs) {
      sh[threadIdx.x] += sh[threadIdx.x + ofs];
      sh2[threadIdx.x] += sh2[threadIdx.x + ofs];
    }
    __syncthreads();
  }
  if (threadIdx.x == 0) {
    float m = sh[0] / (float)M;
    mu[n] = m;
    var[n] = sh2[0] / (float)M - m * m;
  }
}

// ---------------------------------------------------------------------------
// 4) OUT = lrelu(BN(H)) @ W2^T + b2 ; optional fused MSE partial vs REF.
// ---------------------------------------------------------------------------
__global__ void k_gen_out(const float* __restrict__ H, const float* __restrict__ W2,
                          const float* __restrict__ b2, const float* __restrict__ mu,
                          const float* __restrict__ var, const float* __restrict__ gamma,
                          const float* __restrict__ beta, float* __restrict__ OUT,
                          const float* __restrict__ REF, float* __restrict__ msePart,
                          int M, int N, int K) {
  __shared__ float s_scale[HID], s_shift[HID];
  for (int k = threadIdx.x; k < K; k += 32) {
    float sc = gamma[k] * rsqrtf(var[k] + BN_EPS);
    s_scale[k] = sc;
    s_shift[k] = beta[k] - mu[k] * sc;
  }
  __syncthreads();
  int rt = blockIdx.x * 16, ct = blockIdx.y * 16;
  int lane = threadIdx.x & 31, hlf = lane >> 4, lid = lane & 15;
  const float* Arow = H + (rt + lid) * K;
  const float* Brow = W2 + (ct + lid) * K;
  v8f acc = {};
  for (int k = 0; k < K; k += 4) {
    int ka = k + 2 * hlf;
    float h0 = Arow[ka] * s_scale[ka] + s_shift[ka];
    float h1 = Arow[ka + 1] * s_scale[ka + 1] + s_shift[ka + 1];
    v2f a, b;
    a.x = (h0 >= 0.f) ? h0 : LRELU * h0;
    a.y = (h1 >= 0.f) ? h1 : LRELU * h1;
    b.x = Brow[ka]; b.y = Brow[ka + 1];
    acc = __builtin_amdgcn_wmma_f32_16x16x4_f32(false, a, false, b, (short)0, acc,
                                                false, false);
  }
  float lsum = 0.f;
#pragma unroll
  for (int r = 0; r < 8; ++r) {
    int row = rt + r + 8 * hlf, col = ct + lid;
    float v = acc[r] + b2[col];
    OUT[row * N + col] = v;
    if (REF) {
      float d = v - REF[row * N + col];
      lsum += d * d;
    }
  }
  if (REF) {
    for (int m = 16; m > 0; m >>= 1) lsum += __shfl_xor(lsum, m, 32);
    if (lane == 0) msePart[blockIdx.x * gridDim.y + blockIdx.y] = lsum;
  }
  (void)M;
}

// ---------------------------------------------------------------------------
// 5) squared row norms; optional gather through perm
// ---------------------------------------------------------------------------
__global__ void k_rownorm(const float* __restrict__ Z, const int* __restrict__ perm,
                          float* __restrict__ out, int K) {
  int j = blockIdx.x;
  int src = perm ? perm[j] : j;
  const float* row = Z + src * K;
  float s = 0.f;
  for (int k = threadIdx.x; k < K; k += blockDim.x) {
    float v = row[k];
    s += v * v;
  }
  __shared__ float sh[64];
  sh[threadIdx.x] = s;
  __syncthreads();
  for (int ofs = 32; ofs > 0; ofs >>= 1) {
    if (threadIdx.x < ofs) sh[threadIdx.x] += sh[threadIdx.x + ofs];
    __syncthreads();
  }
  if (threadIdx.x == 0) out[j] = sh[0];
}

// ---------------------------------------------------------------------------
// 6) NCT: rowmin over j of (|Zp_i|^2 + |Zs_j|^2 - 2 Zp_i.Zs_j)/LAT.
//    The 16x128 A tile (one contiguous 8KB block of Zp) is DMA'd into LDS by
//    the Tensor Data Mover (tensor_load_to_lds, wave 0 issues, TENSORcnt
//    waited, barrier publishes). Each wave then hoists its A fragments into
//    registers once and runs a fully unrolled 32-WMMA inner loop with a
//    single global b64 load per WMMA (B gathered through perm).
// ---------------------------------------------------------------------------
__global__ void k_nct_min(const float* __restrict__ Zp, const float* __restrict__ zlog,
                          const int* __restrict__ perm, const float* __restrict__ pnorm,
                          const float* __restrict__ znorm, float* __restrict__ rowmin,
                          int Ncols) {
  __shared__ float sA[16 * LAT];
  __shared__ float s_min[8][16];
  int rt = blockIdx.x * 16;

  if (threadIdx.x < 32) {
    // --- TDM descriptor (D#): 1-D tile, 2048 x 4B contiguous ---
    uint64_t ga = (uint64_t)(const void*)(Zp + (size_t)rt * LAT);
    uint32 lds = (uint32)(uintptr_t)(void*)sA;
    v4u g0;
    g0.x = 1u;                                   // count=1, user descriptor
    g0.y = lds;                                  // lds_addr
    g0.z = (uint32)ga;                           // global_addr[31:0]
    g0.w = (uint32)((ga >> 32) & 0x01FFFFFFu) | (2u << 30);  // addr[56:32], type=2
    v8u g1;
    g1[0] = 2u << 16;        // workgroup_mask=0, data_size=2 (4 bytes)
    g1[1] = 2048u << 16;     // tensor_dim0[15:0] = 2048 elements
    g1[2] = 1u << 16;        // tensor_dim0[31:16]=0, tensor_dim1=1
    g1[3] = 2048u << 16;     // tile_dim0 = 2048
    g1[4] = 1u;              // tile_dim1 = 1, tile_dim2 = 0
    g1[5] = 2048u;           // tensor_dim0_stride = 2048
    g1[6] = 0u;
    g1[7] = 0u;
    asm volatile("tensor_load_to_lds %0, %1" :: "s"(g0), "s"(g1) : "memory");
    __builtin_amdgcn_s_wait_tensorcnt(0);
  }
  __syncthreads();

  int wave = threadIdx.x >> 5;
  int lane = threadIdx.x & 31, hlf = lane >> 4, lid = lane & 15;

  // hoist this wave's A fragments from LDS into registers (reused 128x)
  v2f afrag[LAT / 4];
#pragma unroll
  for (int kk = 0; kk < LAT / 4; ++kk) {
    int ka = kk * 4 + 2 * hlf;
    afrag[kk].x = sA[lid * LAT + ka];
    afrag[kk].y = sA[lid * LAT + ka + 1];
  }

  float rmin[8];
#pragma unroll
  for (int r = 0; r < 8; ++r) rmin[r] = 3.4e38f;
  float pn[8];
#pragma unroll
  for (int r = 0; r < 8; ++r) pn[r] = pnorm[rt + r + 8 * hlf];

  for (int ct = wave * 16; ct < Ncols; ct += 8 * 16) {
    int col = ct + lid;
    const float* Brow = zlog + (size_t)perm[col] * LAT;  // Zs[col] gathered
    v8f acc = {};
#pragma unroll
    for (int kk = 0; kk < LAT / 4; ++kk) {
      int ka = kk * 4 + 2 * hlf;
      v2f b;
      b.x = Brow[ka];
      b.y = Brow[ka + 1];
      acc = __builtin_amdgcn_wmma_f32_16x16x4_f32(false, afrag[kk], false, b,
                                                  (short)0, acc, false, false);
    }
    float zn = znorm[col];
#pragma unroll
    for (int r = 0; r < 8; ++r) {
      float l2 = (pn[r] + zn - 2.f * acc[r]) * (1.f / (float)LAT);
      rmin[r] = fminf(rmin[r], l2);
    }
  }
  // min over the 16 lanes of each half (different N columns)
#pragma unroll
  for (int r = 0; r < 8; ++r) {
    float v = rmin[r];
    for (int m = 8; m > 0; m >>= 1) v = fminf(v, __shfl_xor(v, m, 32));
    rmin[r] = v;
  }
  if (lid == 0) {
#pragma unroll
    for (int r = 0; r < 8; ++r) s_min[wave][hlf * 8 + r] = rmin[r];
  }
  __syncthreads();
  if (threadIdx.x < 16) {
    float v = s_min[0][threadIdx.x];
    for (int w = 1; w < 8; ++w) v = fminf(v, s_min[w][threadIdx.x]);
    rowmin[rt + threadIdx.x] = v;
  }
}

// ---------------------------------------------------------------------------
// 7) S = Xc^T Xc  (64x64 gram over Nbatch rows), centered inline with m[].
// ---------------------------------------------------------------------------
__global__ void k_gram(const float* __restrict__ X, const float* __restrict__ m,
                       float* __restrict__ S, int Nbatch, int n) {
  int rt = blockIdx.x * 16, ct = blockIdx.y * 16;
  int lane = threadIdx.x & 31, hlf = lane >> 4, lid = lane & 15;
  float mi = m[rt + lid], mk = m[ct + lid];
  v8f acc = {};
  for (int b = 0; b < Nbatch; b += 4) {
    int kb = b + 2 * hlf;
    v2f a, bb;
    a.x = X[kb * n + rt + lid] - mi;
    a.y = X[(kb + 1) * n + rt + lid] - mi;
    bb.x = X[kb * n + ct + lid] - mk;
    bb.y = X[(kb + 1) * n + ct + lid] - mk;
    acc = __builtin_amdgcn_wmma_f32_16x16x4_f32(false, a, false, bb, (short)0, acc,
                                                false, false);
  }
  wmma_store_tile(S + rt * n + ct, n, acc);
}

// ---------------------------------------------------------------------------
// 8) loss_ind from S, C using the exact closed form:
//    G[j,i,k] = S[i,k] - S[i,j]S[j,k]/s2[j];  dg[j,i] = s2[i] - S[i,j]^2/s2[j]
// ---------------------------------------------------------------------------
__global__ void k_loss_ind(const float* __restrict__ S, const float* __restrict__ C,
                           float* __restrict__ out) {
  __shared__ float sS[4096], sC[4096], s2[64], sh[256];
  for (int i = threadIdx.x; i < 4096; i += 256) {
    sS[i] = S[i];
    sC[i] = C[i];
  }
  __syncthreads();
  if (threadIdx.x < 64) s2[threadIdx.x] = sS[threadIdx.x * 64 + threadIdx.x];
  __syncthreads();
  float acc = 0.f;
  for (int p = threadIdx.x; p < 4096; p += 256) {
    int i = p >> 6, k = p & 63;
    if (i != k) acc += 62.f * (sS[p] * sS[p]) / (s2[i] * s2[k]);
  }
  for (int t = threadIdx.x; t < 64 * 64 * 64; t += 256) {
    int j = t >> 12, i = (t >> 6) & 63, k = t & 63;
    if (i == j || k == j || i == k) continue;
    float sij = sS[i * 64 + j], sjk = sS[j * 64 + k], skj = sS[k * 64 + j];
    float inv = 1.f / s2[j];
    float G = sS[i * 64 + k] - sij * sjk * inv;
    float dgi = s2[i] - sij * sij * inv;
    float dgk = s2[k] - skj * skj * inv;
    float scc = (G * G) / (dgi * dgk);
    float wt = (sC[i * 64 + j] + sC[j * 64 + i]) * sC[j * 64 + k];
    acc += wt * scc;
  }
  sh[threadIdx.x] = acc;
  __syncthreads();
  for (int ofs = 128; ofs > 0; ofs >>= 1) {
    if (threadIdx.x < ofs) sh[threadIdx.x] += sh[threadIdx.x + ofs];
    __syncthreads();
  }
  if (threadIdx.x == 0) *out = sh[0];
}

// ---------------------------------------------------------------------------
// 9) finalize: loss = loss_trans + mse/(NS*SIZE) + mean(rowmin) + loss_ind
// ---------------------------------------------------------------------------
__global__ void k_finalize(const float* __restrict__ accum,
                           const float* __restrict__ msePart, int nMse,
                           const float* __restrict__ rowmin, int nRows,
                           float* __restrict__ out) {
  __shared__ float sh[256];
  int t = threadIdx.x;
  float s = 0.f;
  for (int i = t; i < nMse; i += 256) s += msePart[i];
  sh[t] = s;
  __syncthreads();
  for (int ofs = 128; ofs > 0; ofs >>= 1) {
    if (t < ofs) sh[t] += sh[t + ofs];
    __syncthreads();
  }
  float mse = sh[0] / ((float)NS * (float)SIZE);
  __syncthreads();
  s = 0.f;
  for (int i = t; i < nRows; i += 256) s += rowmin[i];
  sh[t] = s;
  __syncthreads();
  for (int ofs = 128; ofs > 0; ofs >>= 1) {
    if (t < ofs) sh[t] += sh[t + ofs];
    __syncthreads();
  }
  if (t == 0) out[0] = accum[0] + mse + sh[0] / (float)BTR + accum[1];
}

// ===========================================================================
extern "C" void kernel_launch(void* const* d_in, const int* in_sizes, int n_in,
                              void* d_out, int out_size, void* d_ws, size_t ws_size,
                              hipStream_t stream) {
  const float* X          = (const float*)d_in[0];
  const float* noise_tr   = (const float*)d_in[1];
  const float* noise_ind  = (const float*)d_in[2];
  const float* conn       = (const float*)d_in[3];
  const float* zlog       = (const float*)d_in[4];
  const float* glo_W1     = (const float*)d_in[5];
  const float* glo_gamma  = (const float*)d_in[6];
  const float* glo_beta   = (const float*)d_in[7];
  const float* glo_W2     = (const float*)d_in[8];
  const float* glo_b2     = (const float*)d_in[9];
  const float* tr_W1      = (const float*)d_in[10];
  const float* tr_gamma   = (const float*)d_in[11];
  const float* tr_beta    = (const float*)d_in[12];
  const float* tr_W2      = (const float*)d_in[13];
  const float* tr_b2      = (const float*)d_in[14];
  const int*   perm       = (const int*)d_in[15];
  (void)in_sizes; (void)n_in; (void)out_size; (void)ws_size;

  float* ws = (float*)d_ws;
  // ---- workspace layout (floats), ~28.4 MB total ----
  float* accum   = ws;                 // [0]=loss_trans, [1]=loss_ind
  float* wsC     = ws + 16;            // 4096
  float* mu      = wsC + 4096;         // 256
  float* var     = mu + 256;           // 256
  float* m64     = var + 256;          // 64
  float* var64   = m64 + 64;           // 64
  float* S       = var64 + 64;         // 4096
  float* msePart = S + 4096;           // 4096   (1024*4 blocks)
  float* rowmin  = msePart + 4096;     // 2048
  float* pnorm   = rowmin + 2048;      // 2048
  float* znorm   = pnorm + 2048;       // 16384
  float* h       = znorm + 16384;      // NS*HID = 4,194,304 (reused 4x)
  float* Xglo    = h + (size_t)NS * HID;        // NS*SIZE
  float* Zp      = Xglo + (size_t)NS * SIZE;    // BTR*LAT
  float* Zind    = Zp + (size_t)BTR * LAT;      // NIND*LAT
  float* Xind    = Zind + (size_t)NIND * LAT;   // NIND*SIZE

  // 1) connectivity + loss_trans
  k_conn<<<1, 256, 0, stream>>>(conn, wsC, accum + 0);

  // 2) GLO decode of z_logits + fused MSE vs X
  k_gemm_xwT<<<dim3(NS / 16, HID / 16), 32, 0, stream>>>(zlog, glo_W1, h, NS, HID, LAT);
  k_bn_stats<<<HID, 256, 0, stream>>>(h, mu, var, NS, HID);
  k_gen_out<<<dim3(NS / 16, SIZE / 16), 32, 0, stream>>>(
      h, glo_W2, glo_b2, mu, var, glo_gamma, glo_beta, Xglo, X, msePart, NS, SIZE, HID);

  // 3) translator on noise_trans -> Zp
  k_gemm_xwT<<<dim3(BTR / 16, HID / 16), 32, 0, stream>>>(noise_tr, tr_W1, h, BTR, HID, NOISE);
  k_bn_stats<<<HID, 256, 0, stream>>>(h, mu, var, BTR, HID);
  k_gen_out<<<dim3(BTR / 16, LAT / 16), 32, 0, stream>>>(
      h, tr_W2, tr_b2, mu, var, tr_gamma, tr_beta, Zp, nullptr, nullptr, BTR, LAT, HID);

  // 4) translator on noise_indep -> Z_ind
  k_gemm_xwT<<<dim3(NIND / 16, HID / 16), 32, 0, stream>>>(noise_ind, tr_W1, h, NIND, HID, NOISE);
  k_bn_stats<<<HID, 256, 0, stream>>>(h, mu, var, NIND, HID);
  k_gen_out<<<dim3(NIND / 16, LAT / 16), 32, 0, stream>>>(
      h, tr_W2, tr_b2, mu, var, tr_gamma, tr_beta, Zind, nullptr, nullptr, NIND, LAT, HID);

  // 5) GLO decode of Z_ind -> X_ind
  k_gemm_xwT<<<dim3(NIND / 16, HID / 16), 32, 0, stream>>>(Zind, glo_W1, h, NIND, HID, LAT);
  k_bn_stats<<<HID, 256, 0, stream>>>(h, mu, var, NIND, HID);
  k_gen_out<<<dim3(NIND / 16, SIZE / 16), 32, 0, stream>>>(
      h, glo_W2, glo_b2, mu, var, glo_gamma, glo_beta, Xind, nullptr, nullptr, NIND, SIZE, HID);

  // 6) row norms for NCT (Zs gathered via perm)
  k_rownorm<<<BTR, 64, 0, stream>>>(Zp, nullptr, pnorm, LAT);
  k_rownorm<<<NS, 64, 0, stream>>>(zlog, perm, znorm, LAT);

  // 7) NCT min-distance (TDM-staged A, register-hoisted fragments, WMMA)
  k_nct_min<<<BTR / 16, 256, 0, stream>>>(Zp, zlog, perm, pnorm, znorm, rowmin, NS);

  // 8) independence loss: colmean -> centered gram -> closed-form reduction
  k_bn_stats<<<SIZE, 256, 0, stream>>>(Xind, m64, var64, NIND, SIZE);
  k_gram<<<dim3(SIZE / 16, SIZE / 16), 32, 0, stream>>>(Xind, m64, S, NIND, SIZE);
  k_loss_ind<<<1, 256, 0, stream>>>(S, wsC, accum + 1);

  // 9) combine everything into the scalar loss
  k_finalize<<<1, 256, 0, stream>>>(accum, msePart, (NS / 16) * (SIZE / 16),
                                    rowmin, BTR, (float*)d_out);
}